// LightGCN_26036091748913
// MI455X (gfx1250) — compile-verified
//
#include <hip/hip_runtime.h>

#define NUM_USERS 100000
#define NUM_ITEMS 50000
#define N_NODES   150000
#define EMBED_DIM 64
#define NUM_EDGES 2000000

// 16 threads per edge, 4 dims (one float4 / global_load_b128) per thread.
#define THREADS_PER_EDGE 16
#define BLOCK 256

// Hardware FP32 global atomic add (global_atomic_add_f32, no return, no CAS).
__device__ __forceinline__ void atomic_add_f32(float* p, float v) {
    unsafeAtomicAdd(p, v);
}

// ---------------------------------------------------------------------------
// Pass 1: agg = S(x0)   (x0 gathered from split user/item tables)
// Symmetric scatter: agg[row] += x0[col]; agg[col] += x0[row].
// ---------------------------------------------------------------------------
__global__ __launch_bounds__(BLOCK)
void LightGCN_scatter1_kernel(const int*   __restrict__ idx,
                              const float* __restrict__ user,
                              const float* __restrict__ item,
                              float*       __restrict__ agg) {
    const int t = blockIdx.x * BLOCK + threadIdx.x;   // grid sized exactly: no bounds check
    const int e = t >> 4;                              // edge id
    const int d = (t & (THREADS_PER_EDGE - 1)) << 2;   // dim quad start

    // Stream-prefetch the edge list ahead. Locality 3 -> WGP scope:
    // pulls into all cache levels (the consuming loads go through WGP$).
    if ((t & (THREADS_PER_EDGE - 1)) == 0) {
        __builtin_prefetch(idx + e + 4096, 0, 3);
        __builtin_prefetch(idx + NUM_EDGES + e + 4096, 0, 3);
    }

    const int row = idx[e];
    const int col = idx[NUM_EDGES + e];

    const float* xr = (row < NUM_USERS)
        ? (user + (size_t)row * EMBED_DIM)
        : (item + (size_t)(row - NUM_USERS) * EMBED_DIM);
    const float* xc = (col < NUM_USERS)
        ? (user + (size_t)col * EMBED_DIM)
        : (item + (size_t)(col - NUM_USERS) * EMBED_DIM);

    const float4 vc = *(const float4*)(xc + d);  // -> agg[row]
    const float4 vr = *(const float4*)(xr + d);  // -> agg[col]

    float* dr = agg + (size_t)row * EMBED_DIM + d;
    float* dc = agg + (size_t)col * EMBED_DIM + d;

    atomic_add_f32(dr + 0, vc.x);
    atomic_add_f32(dr + 1, vc.y);
    atomic_add_f32(dr + 2, vc.z);
    atomic_add_f32(dr + 3, vc.w);
    atomic_add_f32(dc + 0, vr.x);
    atomic_add_f32(dc + 1, vr.y);
    atomic_add_f32(dc + 2, vr.z);
    atomic_add_f32(dc + 3, vr.w);
}

// ---------------------------------------------------------------------------
// Pass 2: out = 0.25 * S(agg)   (both 0.5 layer scales folded in here)
// ---------------------------------------------------------------------------
__global__ __launch_bounds__(BLOCK)
void LightGCN_scatter2_kernel(const int*   __restrict__ idx,
                              const float* __restrict__ agg,
                              float*       __restrict__ out) {
    const int t = blockIdx.x * BLOCK + threadIdx.x;
    const int e = t >> 4;
    const int d = (t & (THREADS_PER_EDGE - 1)) << 2;

    if ((t & (THREADS_PER_EDGE - 1)) == 0) {
        __builtin_prefetch(idx + e + 4096, 0, 3);
        __builtin_prefetch(idx + NUM_EDGES + e + 4096, 0, 3);
    }

    const int row = idx[e];
    const int col = idx[NUM_EDGES + e];

    const float4 vc = *(const float4*)(agg + (size_t)col * EMBED_DIM + d);
    const float4 vr = *(const float4*)(agg + (size_t)row * EMBED_DIM + d);

    float* dr = out + (size_t)row * EMBED_DIM + d;
    float* dc = out + (size_t)col * EMBED_DIM + d;

    const float s = 0.25f;
    atomic_add_f32(dr + 0, s * vc.x);
    atomic_add_f32(dr + 1, s * vc.y);
    atomic_add_f32(dr + 2, s * vc.z);
    atomic_add_f32(dr + 3, s * vc.w);
    atomic_add_f32(dc + 0, s * vr.x);
    atomic_add_f32(dc + 1, s * vr.y);
    atomic_add_f32(dc + 2, s * vr.z);
    atomic_add_f32(dc + 3, s * vr.w);
}

// ---------------------------------------------------------------------------
// Zero-init both accumulator buffers (harness poisons d_out / d_ws to 0xAA).
// ---------------------------------------------------------------------------
__global__ __launch_bounds__(BLOCK)
void LightGCN_zero_kernel(float4* __restrict__ a, float4* __restrict__ b, int n4) {
    const float4 z = make_float4(0.f, 0.f, 0.f, 0.f);
    int stride = gridDim.x * blockDim.x;
    for (int i = blockIdx.x * blockDim.x + threadIdx.x; i < n4; i += stride) {
        a[i] = z;
        b[i] = z;
    }
}

// ---------------------------------------------------------------------------
// kernel_launch
//   d_in[0] = edge_index  (int32, [2, NUM_EDGES] row-major)
//   d_in[1] = user_emb    (f32, [NUM_USERS, 64])
//   d_in[2] = item_emb    (f32, [NUM_ITEMS, 64])
//   d_out   = [150000, 64] f32
//   d_ws    = layer-1 accumulator, 38.4 MB
// ---------------------------------------------------------------------------
extern "C" void kernel_launch(void* const* d_in, const int* in_sizes, int n_in,
                              void* d_out, int out_size, void* d_ws, size_t ws_size,
                              hipStream_t stream) {
    (void)in_sizes; (void)n_in; (void)out_size; (void)ws_size;

    const int*   idx  = (const int*)d_in[0];
    const float* user = (const float*)d_in[1];
    const float* item = (const float*)d_in[2];
    float*       out  = (float*)d_out;
    float*       agg  = (float*)d_ws;

    const int n_elems = N_NODES * EMBED_DIM;      // 9,600,000
    const int n4      = n_elems / 4;              // 2,400,000 float4s
    const int zero_blocks = (n4 + BLOCK - 1) / BLOCK;

    LightGCN_zero_kernel<<<zero_blocks, BLOCK, 0, stream>>>(
        (float4*)agg, (float4*)out, n4);

    const long long total_threads = (long long)NUM_EDGES * THREADS_PER_EDGE;  // 32M
    const int scatter_blocks = (int)((total_threads + BLOCK - 1) / BLOCK);    // 125000 exact

    LightGCN_scatter1_kernel<<<scatter_blocks, BLOCK, 0, stream>>>(idx, user, item, agg);
    LightGCN_scatter2_kernel<<<scatter_blocks, BLOCK, 0, stream>>>(idx, agg, out);
}